// HamModule_9182640078936
// MI455X (gfx1250) — compile-verified
//
#include <hip/hip_runtime.h>
#include <hip/hip_bf16.h>
#include <math.h>

typedef float v2f __attribute__((ext_vector_type(2)));
typedef float v8f __attribute__((ext_vector_type(8)));

#define LCH 12
#define DIM 4096
#define KB  16    // subspace block width == WMMA N
#define NE  4
#define KSPLIT 8  // K-chunks (waves) per GEMM workgroup

// workspace float offsets
#define OFF_HDR  0              // [0]=sigma, [1..12]=Bz, [13..24]=Bx
#define OFF_G    64             // 16x16 Gram / projected matrix
#define OFF_L    320            // 16x16 Cholesky factor
#define OFF_EVAL 576            // 16 eigenvalues
#define OFF_EVEC 592            // 16x16 eigenvectors (columns)
#define OFF_GP   1024           // 16 partial 16x16 Gram tiles (4096 floats)
#define OFF_X    8192           // 4096x16
#define OFF_Y    (OFF_X + DIM*KB)
#define OFF_H    (OFF_Y + DIM*KB)   // 4096x4096 = sigma*I - H

// ---------------------------------------------------------------------------
// 1) phi profile, fields, Gershgorin shift sigma (single thread, trivial work)
// ---------------------------------------------------------------------------
__global__ void prep_kernel(const float* __restrict__ B0,
                            const float* __restrict__ Bext,
                            const float* __restrict__ pd,
                            float* __restrict__ hdr) {
  if (threadIdx.x != 0 || blockIdx.x != 0) return;
  float cum[6];
  float s = 0.f;
  for (int i = 0; i < 6; ++i) { s += pd[i] * pd[i]; cum[i] = s; }
  float phi6[6];
  for (int i = 0; i < 6; ++i) phi6[i] = cum[i] * 3.14159265358979f / cum[5];
  float p0 = phi6[0];
  for (int i = 0; i < 6; ++i) phi6[i] -= p0;
  float phi[LCH];
  for (int i = 0; i < 6; ++i) { phi[i] = phi6[i]; phi[11 - i] = phi6[i]; }
  float b0 = B0[0], be = Bext[0];
  float sabsz = 0.f, sabsx = 0.f;
  for (int i = 0; i < LCH; ++i) {
    float bz = b0 * cosf(phi[i]) + be;
    float bx = b0 * sinf(phi[i]);
    hdr[1 + i]  = bz;
    hdr[13 + i] = bx;
    sabsz += fabsf(bz);
    sabsx += fabsf(bx);
  }
  // Gershgorin upper bound on lambda_max of H, plus safety margin
  float sigma = 0.25f * 11.f + 0.5f * sabsz + 0.5f * 11.f + 0.5f * sabsx + 1.0f;
  hdr[0] = sigma;
}

// ---------------------------------------------------------------------------
// 2) dense build of A = sigma*I - H  (one thread per matrix entry)
// ---------------------------------------------------------------------------
__global__ void build_h_kernel(float* __restrict__ Hp, const float* __restrict__ hdr) {
  size_t idx = (size_t)blockIdx.x * blockDim.x + threadIdx.x;
  int r = (int)(idx >> 12);
  int c = (int)(idx & (DIM - 1));
  unsigned x = (unsigned)(r ^ c);
  float val = 0.f;
  if (x == 0u) {
    // diagonal: J1 * sum Sz_i Sz_{i+1} + sum Sz_i * Bz_i
    float d = 0.f;
    for (int i = 0; i < LCH; ++i) {
      float szi = 0.5f - (float)((r >> i) & 1);
      d += szi * hdr[1 + i];
      if (i < LCH - 1) {
        float szn = 0.5f - (float)((r >> (i + 1)) & 1);
        d += szi * szn;
      }
    }
    val = d;
  } else if ((x & (x - 1u)) == 0u) {
    // single spin flip: transverse field Sx_i -> Bx_i / 2
    int i = __ffs((int)x) - 1;
    if (i < LCH) val = 0.5f * hdr[13 + i];
  } else {
    // adjacent pair flip-flop: 0.5*J1 when bits differ
    int i = __ffs((int)x) - 1;
    if (i < LCH - 1 && x == (3u << i)) {
      int bi = (r >> i) & 1, bj = (r >> (i + 1)) & 1;
      if (bi != bj) val = 0.5f;
    }
  }
  float sig = hdr[0];
  Hp[idx] = (x == 0u ? sig : 0.f) - val;
}

// ---------------------------------------------------------------------------
// 3) deterministic pseudo-random init of the subspace block
// ---------------------------------------------------------------------------
__device__ __forceinline__ unsigned wang(unsigned x) {
  x = (x ^ 61u) ^ (x >> 16); x *= 9u; x ^= x >> 4;
  x *= 0x27d4eb2du; x ^= x >> 15; return x;
}
__global__ void init_x_kernel(float* __restrict__ Y) {
  int t = blockIdx.x * blockDim.x + threadIdx.x;   // DIM*KB
  unsigned h = wang((unsigned)t + 0x9e3779b9u);
  Y[t] = ((float)(h & 0xFFFFu) - 32768.f) * (1.f / 32768.f);
}

// ---------------------------------------------------------------------------
// 4) Y = A * X via V_WMMA_F32_16X16X4_F32.
//    8 waves per block, each wave reduces a 512-wide K-chunk into its own
//    accumulator tile; wave 0 sums the 8 partials from LDS in fixed order
//    (deterministic). 2048 waves total -> enough latency hiding for the
//    L2-resident streaming of the 64MB matrix.
// ---------------------------------------------------------------------------
__global__ void gemm_kernel(const float* __restrict__ Hp,
                            const float* __restrict__ X,
                            float* __restrict__ Y) {
  __shared__ float part[KSPLIT][32][8];
  const int row0 = blockIdx.x * 16;
  const int tid  = threadIdx.x;      // 256 threads = 8 waves
  const int wave = tid >> 5;
  const int lane = tid & 31;
  const int half = lane >> 4;        // 0: K={0,1}, 1: K={2,3}
  const int l    = lane & 15;        // A: row M=l ; B/C: col N=l
  const int k0   = wave * (DIM / KSPLIT);

  v8f c = {};
  const float* arow = Hp + (size_t)(row0 + l) * DIM + (half << 1);

  for (int k = k0; k < k0 + DIM / KSPLIT; k += 4) {
    // A fragment (16x4): lane<16 holds (K=0,1) of row l; lane>=16 holds (K=2,3)
    float2 av = *(const float2*)(arow + k);
    v2f a; a.x = av.x; a.y = av.y;
    // B fragment (4x16): VGPR0 = rows K=0/K=2 striped over lanes, VGPR1 = K=1/K=3
    v2f b;
    b.x = X[(k + (half << 1)    ) * KB + l];
    b.y = X[(k + (half << 1) + 1) * KB + l];
    c = __builtin_amdgcn_wmma_f32_16x16x4_f32(false, a, false, b,
                                              (short)0, c, false, false);
  }

#pragma unroll
  for (int v = 0; v < 8; ++v) part[wave][lane][v] = c[v];
  __syncthreads();

  if (wave == 0) {
    // C/D layout: VGPR v -> M=v (lanes 0-15), M=v+8 (lanes 16-31), N=lane&15
    float* yout = Y + (size_t)row0 * KB;
#pragma unroll
    for (int v = 0; v < 8; ++v) {
      float s = 0.f;
#pragma unroll
      for (int w = 0; w < KSPLIT; ++w) s += part[w][lane][v];
      yout[(v + (half << 3)) * KB + l] = s;
    }
  }
}

// ---------------------------------------------------------------------------
// 5) G = A^T B (16x16): two-stage deterministic reduction.
//    Stage 1: 16 blocks, each reduces 256 rows -> partial tile.
//    Stage 2: 1 block sums the 16 partials in fixed order.
// ---------------------------------------------------------------------------
__global__ void gram_part_kernel(const float* __restrict__ A,
                                 const float* __restrict__ B,
                                 float* __restrict__ Gp) {
  int t = threadIdx.x;          // 256 threads = 16x16 entries
  int i = t >> 4, j = t & 15;
  int r0 = blockIdx.x * (DIM / 16);
  float s = 0.f;
  for (int r = r0; r < r0 + DIM / 16; ++r)
    s = fmaf(A[r * KB + i], B[r * KB + j], s);
  Gp[blockIdx.x * 256 + t] = s;
}

__global__ void gram_reduce_kernel(const float* __restrict__ Gp,
                                   float* __restrict__ G) {
  int t = threadIdx.x;          // 256
  float s = 0.f;
#pragma unroll
  for (int b = 0; b < 16; ++b) s += Gp[b * 256 + t];
  G[t] = s;
}

// ---------------------------------------------------------------------------
// 6) 16x16 Cholesky G = L L^T (single thread)
// ---------------------------------------------------------------------------
__global__ void chol_kernel(const float* __restrict__ G, float* __restrict__ Lc) {
  if (threadIdx.x || blockIdx.x) return;
  float a[KB][KB];
  for (int i = 0; i < KB; ++i)
    for (int j = 0; j < KB; ++j) a[i][j] = 0.f;
  for (int j = 0; j < KB; ++j) {
    float s = G[j * KB + j];
    for (int k = 0; k < j; ++k) s -= a[j][k] * a[j][k];
    float d = sqrtf(fmaxf(s, 1e-12f));
    a[j][j] = d;
    float inv = 1.f / d;
    for (int i = j + 1; i < KB; ++i) {
      float t2 = G[i * KB + j];
      for (int k = 0; k < j; ++k) t2 -= a[i][k] * a[j][k];
      a[i][j] = t2 * inv;
    }
  }
  for (int i = 0; i < KB; ++i)
    for (int j = 0; j < KB; ++j) Lc[i * KB + j] = a[i][j];
}

// ---------------------------------------------------------------------------
// 7) X = Y * L^-T  (row-wise forward substitution; Cholesky-QR)
// ---------------------------------------------------------------------------
__global__ void apply_linv_kernel(const float* __restrict__ src,
                                  const float* __restrict__ Lc,
                                  float* __restrict__ dst) {
  __shared__ float Ls[KB * KB];
  int t = threadIdx.x;
  if (t < KB * KB) Ls[t] = Lc[t];
  __syncthreads();
  int r = blockIdx.x * blockDim.x + t;
  float y[KB], xv[KB];
#pragma unroll
  for (int j = 0; j < KB; ++j) y[j] = src[r * KB + j];
  for (int j = 0; j < KB; ++j) {
    float s = y[j];
    for (int k = 0; k < j; ++k) s -= Ls[j * KB + k] * xv[k];
    xv[j] = s / Ls[j * KB + j];
  }
#pragma unroll
  for (int j = 0; j < KB; ++j) dst[r * KB + j] = xv[j];
}

// ---------------------------------------------------------------------------
// 8) 16x16 symmetric Jacobi eigensolver; eigenvalues sorted DESC (single thread)
// ---------------------------------------------------------------------------
__global__ void jacobi_kernel(const float* __restrict__ G,
                              float* __restrict__ Eval,
                              float* __restrict__ Evec) {
  if (threadIdx.x || blockIdx.x) return;
  float a[KB][KB], V[KB][KB];
  for (int i = 0; i < KB; ++i)
    for (int j = 0; j < KB; ++j) {
      a[i][j] = 0.5f * (G[i * KB + j] + G[j * KB + i]);
      V[i][j] = (i == j) ? 1.f : 0.f;
    }
  for (int sweep = 0; sweep < 30; ++sweep) {
    for (int p = 0; p < KB - 1; ++p)
      for (int q = p + 1; q < KB; ++q) {
        float apq = a[p][q];
        if (fabsf(apq) < 1e-12f) continue;
        float theta = (a[q][q] - a[p][p]) / (2.f * apq);
        float t = (theta >= 0.f ? 1.f : -1.f) /
                  (fabsf(theta) + sqrtf(1.f + theta * theta));
        float cc = 1.f / sqrtf(1.f + t * t);
        float ss = t * cc;
        for (int k = 0; k < KB; ++k) {
          float akp = a[k][p], akq = a[k][q];
          a[k][p] = cc * akp - ss * akq;
          a[k][q] = ss * akp + cc * akq;
        }
        for (int k = 0; k < KB; ++k) {
          float apk = a[p][k], aqk = a[q][k];
          a[p][k] = cc * apk - ss * aqk;
          a[q][k] = ss * apk + cc * aqk;
        }
        for (int k = 0; k < KB; ++k) {
          float vkp = V[k][p], vkq = V[k][q];
          V[k][p] = cc * vkp - ss * vkq;
          V[k][q] = ss * vkp + cc * vkq;
        }
      }
  }
  float d[KB]; int ord[KB];
  for (int i = 0; i < KB; ++i) { d[i] = a[i][i]; ord[i] = i; }
  for (int i = 0; i < KB; ++i)
    for (int j = i + 1; j < KB; ++j)
      if (d[j] > d[i]) {
        float td = d[i]; d[i] = d[j]; d[j] = td;
        int to = ord[i]; ord[i] = ord[j]; ord[j] = to;
      }
  for (int j = 0; j < KB; ++j) Eval[j] = d[j];
  for (int i = 0; i < KB; ++i)
    for (int j = 0; j < KB; ++j) Evec[i * KB + j] = V[i][ord[j]];
}

// ---------------------------------------------------------------------------
// 9) Ritz vectors V = X * S[:, :4], eigenvalues w = sigma - d  -> d_out
// ---------------------------------------------------------------------------
__global__ void ritz_kernel(const float* __restrict__ X,
                            const float* __restrict__ Evec,
                            const float* __restrict__ Eval,
                            const float* __restrict__ hdr,
                            float* __restrict__ out) {
  int r = blockIdx.x * blockDim.x + threadIdx.x;   // 0..4095
  if (r == 0) {
    for (int j = 0; j < NE; ++j) out[j] = hdr[0] - Eval[j];
  }
  float xr[KB];
#pragma unroll
  for (int k = 0; k < KB; ++k) xr[k] = X[r * KB + k];
  for (int j = 0; j < NE; ++j) {
    float s = 0.f;
#pragma unroll
    for (int k = 0; k < KB; ++k) s = fmaf(xr[k], Evec[k * KB + j], s);
    out[NE + r * NE + j] = s;   // v[:, :4] row-major, matches jnp flatten
  }
}

// ---------------------------------------------------------------------------
extern "C" void kernel_launch(void* const* d_in, const int* in_sizes, int n_in,
                              void* d_out, int out_size, void* d_ws, size_t ws_size,
                              hipStream_t stream) {
  const float* B0   = (const float*)d_in[0];
  const float* Bext = (const float*)d_in[1];
  const float* pd   = (const float*)d_in[2];
  float* ws   = (float*)d_ws;
  float* hdr  = ws + OFF_HDR;
  float* G    = ws + OFF_G;
  float* Lc   = ws + OFF_L;
  float* Eval = ws + OFF_EVAL;
  float* Evec = ws + OFF_EVEC;
  float* Gp   = ws + OFF_GP;
  float* X    = ws + OFF_X;
  float* Y    = ws + OFF_Y;
  float* Hp   = ws + OFF_H;
  float* out  = (float*)d_out;

  prep_kernel<<<1, 32, 0, stream>>>(B0, Bext, pd, hdr);
  build_h_kernel<<<(DIM * DIM) / 256, 256, 0, stream>>>(Hp, hdr);

  // random block -> Cholesky-QR orthonormalization into X
  init_x_kernel<<<(DIM * KB) / 256, 256, 0, stream>>>(Y);
  gram_part_kernel<<<16, 256, 0, stream>>>(Y, Y, Gp);
  gram_reduce_kernel<<<1, 256, 0, stream>>>(Gp, G);
  chol_kernel<<<1, 1, 0, stream>>>(G, Lc);
  apply_linv_kernel<<<DIM / 256, 256, 0, stream>>>(Y, Lc, X);

  // block power iteration on A = sigma*I - H; re-orthonormalize every 3 matvecs
  const int NOUTER = 85;
  for (int it = 0; it < NOUTER; ++it) {
    gemm_kernel<<<DIM / 16, 256, 0, stream>>>(Hp, X, Y);
    gemm_kernel<<<DIM / 16, 256, 0, stream>>>(Hp, Y, X);
    gemm_kernel<<<DIM / 16, 256, 0, stream>>>(Hp, X, Y);
    gram_part_kernel<<<16, 256, 0, stream>>>(Y, Y, Gp);
    gram_reduce_kernel<<<1, 256, 0, stream>>>(Gp, G);
    chol_kernel<<<1, 1, 0, stream>>>(G, Lc);
    apply_linv_kernel<<<DIM / 256, 256, 0, stream>>>(Y, Lc, X);
  }

  // Rayleigh-Ritz: P = X^T A X, eig(P), lowest-4 of H = sigma - top-4 of P
  gemm_kernel<<<DIM / 16, 256, 0, stream>>>(Hp, X, Y);
  gram_part_kernel<<<16, 256, 0, stream>>>(X, Y, Gp);
  gram_reduce_kernel<<<1, 256, 0, stream>>>(Gp, G);
  jacobi_kernel<<<1, 1, 0, stream>>>(G, Eval, Evec);
  ritz_kernel<<<DIM / 256, 256, 0, stream>>>(X, Evec, Eval, hdr, out);
}